// CSPN_43404939493937
// MI455X (gfx1250) — compile-verified
//
#include <hip/hip_runtime.h>

// CSPN for MI455X / gfx1250 (wave32).
// Phase 1: 3x3 conv 32->8 as a WMMA f32 GEMM (M=8 padded to 16, K=288,
//          N=B*H*W) fused with the affinity normalization -> 9 kernel planes.
//          Zero-padding handled branchlessly (clamped address + float mask)
//          so the wave never leaves full-EXEC in the WMMA loop; per-tap loads
//          are batched ahead of the 8 WMMAs for latency hiding.
// Phase 2: 12 memory-bound propagation iterations, 4 pixels/thread with
//          b128 kernel-plane loads; kernel planes (44 MB) stay L2-resident.

typedef __attribute__((ext_vector_type(2))) float v2f;
typedef __attribute__((ext_vector_type(4))) float v4f;
typedef __attribute__((ext_vector_type(8))) float v8f;

constexpr int BATCH = 4, CIN = 32, H = 480, W = 640;
constexpr int HW   = H * W;
constexpr int NPIX = BATCH * HW;          // 1,228,800
constexpr int KTOT = CIN * 9;             // 288 (K ordered k = tap*32 + c)
constexpr int LDS_STRIDE = 292;           // padded A row stride: 8B-aligned,
                                          // 292%64=36 -> 16 distinct banks

// ---------------------------------------------------------------------------
// Kernel 1: conv + gen_kernel. One wave32 computes a 16-pixel strip.
// WMMA f32 16x16x4: A = weights (M=out ch, K), B = im2col (K, N=pixel).
// Layout assumptions (ISA 7.12.2):
//   A: lane L holds M=L&15; its 2 VGPRs hold K = k0+2*(L>>4) + {0,1}
//   B: lane L holds N=L&15; its 2 VGPRs hold K = k0+2*(L>>4) + {0,1}
//   C/D: VGPR r, lanes 0-15 -> (M=r, N=lane) => channel r of pixel `lane`
// ---------------------------------------------------------------------------
__global__ __launch_bounds__(256)
void cspn_conv_gen_kernel(const float* __restrict__ kernel_x,
                          const float* __restrict__ W_aff,
                          const float* __restrict__ b_aff,
                          float* __restrict__ kbuf) {
    __shared__ float Aw[16 * LDS_STRIDE];

    const int tid = threadIdx.x;
    // Stage weights into LDS as Aw[m][k], k = tap*32 + c; rows m>=8 are zero.
    for (int idx = tid; idx < 16 * KTOT; idx += 256) {
        const int m = idx / KTOT;
        const int k = idx - m * KTOT;
        float v = 0.0f;
        if (m < 8) {
            const int t  = k >> 5;       // tap 0..8
            const int c  = k & 31;       // input channel
            const int di = t / 3, dj = t - 3 * (t / 3);
            v = W_aff[((m * CIN + c) * 3 + di) * 3 + dj];
        }
        Aw[m * LDS_STRIDE + k] = v;
    }
    __syncthreads();

    const int lane = tid & 31;
    const int wave = tid >> 5;
    const int tile = blockIdx.x * 8 + wave;     // 16 pixels per tile
    const int p0   = tile * 16;

    const int b   = p0 / HW;
    const int rem = p0 - b * HW;
    const int y   = rem / W;
    const int x0  = rem - y * W;                // W%16==0: strip stays in-row

    const int hi = lane >> 4;                   // 0: K pair {0,1}; 1: {2,3}
    const int ln = lane & 15;
    const int xl = x0 + ln;                     // this lane's pixel column (B/N)

    const float* __restrict__ xb = kernel_x + (size_t)b * CIN * HW;

    v8f acc = {};

    for (int t = 0; t < 9; ++t) {               // tap loop (K outer)
        const int di = t / 3, dj = t - 3 * (t / 3);
        const int yy = y + di - 1;
        const int xx = xl + dj - 1;
        const bool valid = (yy >= 0) & (yy < H) & (xx >= 0) & (xx < W);
        // Branchless zero padding: clamp OOB lanes to a safe in-bounds
        // address and multiply loaded values by a 0/1 mask. EXEC stays
        // all-ones through the entire WMMA loop.
        const float msk = valid ? 1.0f : 0.0f;
        const long  pofs = valid ? ((long)yy * W + xx) : 0;
        const long  cb   = pofs + (long)(2 * hi) * HW;   // this lane's K-pair base

        // Batch all 16 channel loads for this tap, then run the 8 WMMAs.
        float bx[8], by[8];
#pragma unroll
        for (int cs = 0; cs < 8; ++cs) {
            bx[cs] = xb[cb + (long)(cs * 4) * HW];
            by[cs] = xb[cb + (long)(cs * 4) * HW + HW];
        }

        const int kbase = t * 32;
#pragma unroll
        for (int cs = 0; cs < 8; ++cs) {
            const int kk = kbase + cs * 4 + 2 * hi;
            v2f a;
            a.x = Aw[ln * LDS_STRIDE + kk];
            a.y = Aw[ln * LDS_STRIDE + kk + 1];
            v2f bv;
            bv.x = bx[cs] * msk;
            bv.y = by[cs] * msk;
            // D = A*B + C, fp32 WMMA (args: neg_a,A,neg_b,B,c_mod,C,reuse_a,reuse_b)
            acc = __builtin_amdgcn_wmma_f32_16x16x4_f32(
                false, a, false, bv, (short)0, acc, false, false);
        }
    }

    // Epilogue: lanes 0..15 own pixel p0+lane; acc[r] = conv channel r.
    // gen_kernel: a = aff/sum|aff|, s = sum a, kernel = [1-s, a...]
    if (lane < 16) {
        float vals[8];
        float abs_sum = 0.0f;
#pragma unroll
        for (int r = 0; r < 8; ++r) {
            vals[r] = acc[r] + b_aff[r];
            abs_sum += __builtin_fabsf(vals[r]);
        }
        const float inv = 1.0f / abs_sum;
        float s = 0.0f;
#pragma unroll
        for (int r = 0; r < 8; ++r) {
            vals[r] *= inv;
            s += vals[r];
        }
        float* __restrict__ op =
            kbuf + (size_t)b * 9 * HW + (size_t)y * W + (x0 + lane);
        op[0] = 1.0f - s;
#pragma unroll
        for (int r = 0; r < 8; ++r)
            op[(size_t)(r + 1) * HW] = vals[r];
    }
}

// ---------------------------------------------------------------------------
// Kernel 2: one propagation iteration, 4 pixels per thread.
// out[p] = sum_k kernel[b,k,y,x] * x[y - oi_k, x - oj_k]  (zero padded)
// OFFSETS order from the reference: product([0,1,-1], repeat=2)
// Branchless padding: rows/edges clamped + mask-multiplied.
// ---------------------------------------------------------------------------
__global__ __launch_bounds__(256)
void cspn_prop_kernel(const float* __restrict__ kbuf,
                      const float* __restrict__ xin,
                      float* __restrict__ xout) {
    const int q   = blockIdx.x * 256 + threadIdx.x;   // quad (4-pixel) index
    const int b   = q / (HW / 4);
    const int rem = q - b * (HW / 4);
    const int y   = rem / (W / 4);
    const int x4  = (rem - y * (W / 4)) * 4;          // W%4==0: quad in-row

    const float* __restrict__ xb = xin + (size_t)b * HW;

    // r[dy+1][i] = x[y+dy][x4-1+i] with zero padding, i in 0..5
    float r[3][6];
#pragma unroll
    for (int dy = -1; dy <= 1; ++dy) {
        const int  yy = y + dy;
        const float rm = (yy >= 0 && yy < H) ? 1.0f : 0.0f;
        const int  yc = yy < 0 ? 0 : (yy > H - 1 ? H - 1 : yy);
        const float* __restrict__ row = xb + (size_t)yc * W;
        const v4f m = *(const v4f*)(row + x4);        // 16B-aligned
        r[dy + 1][1] = m[0] * rm;
        r[dy + 1][2] = m[1] * rm;
        r[dy + 1][3] = m[2] * rm;
        r[dy + 1][4] = m[3] * rm;
        const float lm = (x4 > 0)       ? rm : 0.0f;
        const float qm = (x4 + 4 < W)   ? rm : 0.0f;
        r[dy + 1][0] = row[x4 > 0 ? x4 - 1 : 0] * lm;
        r[dy + 1][5] = row[x4 + 4 < W ? x4 + 4 : W - 1] * qm;
    }

    const float* __restrict__ kp =
        kbuf + (size_t)b * 9 * HW + (size_t)y * W + x4;
    const int OI[9] = {0, 0, 0, 1, 1, 1, -1, -1, -1};
    const int OJ[9] = {0, 1, -1, 0, 1, -1, 0, 1, -1};

    v4f acc = {};
#pragma unroll
    for (int k = 0; k < 9; ++k) {
        const v4f kv = *(const v4f*)(kp + (size_t)k * HW);  // b128, 16B-aligned
#pragma unroll
        for (int j = 0; j < 4; ++j)
            acc[j] += kv[j] * r[1 - OI[k]][1 + j - OJ[k]];
    }
    *(v4f*)(xout + (size_t)q * 4) = acc;
}

// ---------------------------------------------------------------------------
extern "C" void kernel_launch(void* const* d_in, const int* in_sizes, int n_in,
                              void* d_out, int out_size, void* d_ws, size_t ws_size,
                              hipStream_t stream) {
    const float* kernel_x = (const float*)d_in[0];   // [4,32,480,640]
    const float* x        = (const float*)d_in[1];   // [4,1,480,640]
    const float* W_aff    = (const float*)d_in[2];   // [8,32,3,3]
    const float* b_aff    = (const float*)d_in[3];   // [8]
    float* out  = (float*)d_out;                     // [4,1,480,640]
    float* kbuf = (float*)d_ws;                      // 9*NPIX floats
    float* xtmp = kbuf + (size_t)9 * NPIX;           // NPIX floats (ping buffer)

    // Phase 1: conv + kernel generation. 16 px per wave, 8 waves per block.
    cspn_conv_gen_kernel<<<NPIX / (16 * 8), 256, 0, stream>>>(
        kernel_x, W_aff, b_aff, kbuf);

    // Phase 2: 12 iterations, ping-pong so the last write lands in d_out:
    // x -> xtmp -> out -> xtmp -> ... -> out   (odd iters write out)
    const float* cur = x;
    for (int it = 0; it < 12; ++it) {
        float* dst = (it & 1) ? out : xtmp;
        cspn_prop_kernel<<<(NPIX / 4) / 256, 256, 0, stream>>>(kbuf, cur, dst);
        cur = dst;
    }
}